// MoveCenterPts_19318762897440
// MI455X (gfx1250) — compile-verified
//
#include <hip/hip_runtime.h>

// MoveCenterPts for MI455X (gfx1250, wave32).
// out[t,d] = (MOD[d] && f!=0) ? clip(f + deltas[t]·coef[d], 0, LIM[d]) : f
// Memory-bound: ~444 MB streamed @ 23.3 TB/s => ~19 us floor.
// Offsets computed exactly with two chained V_WMMA_F32_16X16X4_F32 (K=8).
// Epilogue: interior tiles (99.2%) are branch-free, 8 loads in flight via
// immediate-offset clause; only the last t/d tile rows take the guarded path.

typedef __attribute__((ext_vector_type(2))) float v2f;
typedef __attribute__((ext_vector_type(8))) float v8f;

#define IM_W_F 1280.0f
#define IM_H_F 720.0f
#define NT 25000
#define ND 2217

// ---- static layout, derived analytically (replays the reference's `ind` walk) ----
// descriptor bits: [7:0] = delta mask, [8] = modified slot, [9] = limit is IM_H
__device__ __forceinline__ unsigned mk2(unsigned a, unsigned b, unsigned isH) {
  return (1u << a) | (1u << b) | 0x100u | (isH << 9);
}
__device__ __forceinline__ unsigned mk1(unsigned a, unsigned isH) {
  return (1u << a) | 0x100u | (isH << 9);
}

__device__ __forceinline__ unsigned feat_desc(int d) {
  enum { RHX = 0, RHY, LHX, LHY, OX, OY, JX, JY };
  if (d == 0 || d == 83 || d == 168) return 0u;               // activation slots
  if (d <= 80)  return ((d - 1) & 1)  ? mk2(RHY, OY, 1) : mk2(RHX, OX, 0);   // obj<->rh
  if (d == 81)  return mk1(RHX, 0);                            // rh center x
  if (d == 82)  return mk1(RHY, 1);                            // rh center y
  if (d <= 163) return ((d - 84) & 1) ? mk2(LHY, OY, 1) : mk2(LHX, OX, 0);   // obj<->lh
  if (d == 164) return mk1(LHX, 0);
  if (d == 165) return mk1(LHY, 1);
  if (d == 166) return mk2(RHX, LHX, 0);                       // rh<->lh
  if (d == 167) return mk2(RHY, LHY, 1);
  if (d <= 368) {                                              // 40 x {act,int,int,ox,oy}
    int r = (d - 169) % 5;
    return (r == 3) ? mk1(OX, 0) : ((r == 4) ? mk1(OY, 1) : 0u);
  }
  if (d <= 412) return ((d - 369) & 1) ? mk2(LHY, JY, 1) : mk2(LHX, JX, 0);  // joint-lh
  if (d <= 456) return ((d - 413) & 1) ? mk2(RHY, JY, 1) : mk2(RHX, JX, 0);  // joint-rh
  return ((d - 457) & 1) ? mk2(OY, JY, 1) : mk2(OX, JX, 0);                  // joint-obj
}

__global__ __launch_bounds__(256)
void move_center_pts_wmma(const float* __restrict__ features,
                          const float* __restrict__ deltas,
                          float* __restrict__ out) {
  constexpr int TILES_D = (ND + 15) / 16;  // 139
  constexpr int TILES_T = (NT + 15) / 16;  // 1563

  const int lane = threadIdx.x & 31;       // gfx1250 is wave32
  const int wave = threadIdx.x >> 5;
  const int tile = blockIdx.x * 8 + wave;  // one 16x16 (t x d) tile per wave
  if (tile >= TILES_D * TILES_T) return;   // wave-uniform exit (EXEC stays all-1)
  const int tile_t = tile / TILES_D;
  const int tile_d = tile - tile_t * TILES_D;
  const int t0 = tile_t * 16;
  const int d0 = tile_d * 16;

  const int n   = lane & 15;   // A: row M ; B: col N ; C/D: col N
  const int hi  = lane >> 4;   // K/M half selector
  const int khi = hi * 2;      // A/B VGPR0/1 hold K = khi, khi+1 (per 16x4 f32 layout)

  // ---- B fragment: 4x16 slice of COEF^T, values in {0,1} from descriptor
  const int dcol = d0 + n;
  const unsigned desc = (dcol < ND) ? feat_desc(dcol) : 0u;
  v2f b0, b1;
  b0.x = (float)((desc >> (khi + 0)) & 1u);  // K-slice 0..3
  b0.y = (float)((desc >> (khi + 1)) & 1u);
  b1.x = (float)((desc >> (khi + 4)) & 1u);  // K-slice 4..7
  b1.y = (float)((desc >> (khi + 5)) & 1u);

  // ---- A fragment: 16x4 slice of deltas (rows = frames), zero-fill past T
  const int tA = t0 + n;
  v2f a0 = {0.0f, 0.0f}, a1 = {0.0f, 0.0f};
  if (tA < NT) {
    const float* dr = deltas + (size_t)tA * 8;
    a0.x = dr[khi + 0];
    a0.y = dr[khi + 1];
    a1.x = dr[khi + 4];
    a1.y = dr[khi + 5];
  }

  // offset tile = A(16x8) x B(8x16) via two chained K=4 f32 WMMAs (exact f32 math)
  v8f c = {};
  c = __builtin_amdgcn_wmma_f32_16x16x4_f32(false, a0, false, b0,
                                            (short)0, c, false, false);
  c = __builtin_amdgcn_wmma_f32_16x16x4_f32(false, a1, false, b1,
                                            (short)0, c, false, false);

  // ---- elementwise epilogue: stream features/out non-temporally (read/write once)
  const bool  modf = (desc >> 8) & 1u;
  const float lim  = ((desc >> 9) & 1u) ? IM_H_F : IM_W_F;
  const int   rbase = t0 + hi * 8;                 // C/D: VGPR i <-> row i + 8*hi

  if (t0 + 16 <= NT && d0 + 16 <= ND) {
    // Interior tile (99.2% of tiles): branch-free. One base address; the
    // i*ND offsets are compile-time constants -> immediate-offset load clause,
    // single wait, then a store clause. 8 loads in flight per wave.
    const size_t base = (size_t)rbase * ND + dcol;
    const float* __restrict__ fp = features + base;
    float* __restrict__       op = out + base;
    float f[8];
#pragma unroll
    for (int i = 0; i < 8; ++i)
      f[i] = __builtin_nontemporal_load(fp + i * ND);
#pragma unroll
    for (int i = 0; i < 8; ++i) {
      float s = fminf(fmaxf(f[i] + c[i], 0.0f), lim);
      const float r = (modf && (f[i] != 0.0f)) ? s : f[i];
      __builtin_nontemporal_store(r, op + i * ND);
    }
  } else {
    // Boundary tile: per-element guards.
    if (dcol >= ND) return;
#pragma unroll
    for (int i = 0; i < 8; ++i) {
      const int t = rbase + i;
      if (t < NT) {
        const size_t idx = (size_t)t * ND + dcol;
        const float f = __builtin_nontemporal_load(features + idx);
        float s = fminf(fmaxf(f + c[i], 0.0f), lim);
        const float r = (modf && (f != 0.0f)) ? s : f;
        __builtin_nontemporal_store(r, out + idx);
      }
    }
  }
}

extern "C" void kernel_launch(void* const* d_in, const int* in_sizes, int n_in,
                              void* d_out, int out_size, void* d_ws, size_t ws_size,
                              hipStream_t stream) {
  (void)in_sizes; (void)n_in; (void)out_size; (void)d_ws; (void)ws_size;
  const float* features = (const float*)d_in[0];  // [25000, 2217] f32
  const float* deltas   = (const float*)d_in[1];  // [25000, 8]    f32
  float* out = (float*)d_out;                     // [25000, 2217] f32

  constexpr int TILES_D = (ND + 15) / 16;            // 139
  constexpr int TILES_T = (NT + 15) / 16;            // 1563
  constexpr int TOTAL   = TILES_D * TILES_T;         // 217257 tiles
  constexpr int BLOCKS  = (TOTAL + 7) / 8;           // 8 waves (tiles) per 256-thr block

  move_center_pts_wmma<<<BLOCKS, 256, 0, stream>>>(features, deltas, out);
}